// HeteroGNN_42666205119271
// MI455X (gfx1250) — compile-verified
//
#include <hip/hip_runtime.h>
#include <hip/hip_bf16.h>
#include <stdint.h>

// ---------------------------------------------------------------------------
// HeteroGNN (SAGEConv, mean aggregation) for MI455X / gfx1250 (wave32, WMMA)
//
// Pipeline:
//   K1: h_c = relu(x_content @ Wc + bc)     WMMA f32 16x16x4, K=256
//   K2: h_u = relu(x_user   @ Wu + bu)      WMMA f32 16x16x4, K=128
//   K3: zero agg/cnt
//   K4: edge scatter: agg[dst] += h_u[src], cnt[dst] += 1   (L2 atomics)
//   K5: fused: mean = agg/max(cnt,1); h = mean@Wl + h_c@Wr + bl; out = h@Wo+bo
// ---------------------------------------------------------------------------

typedef __attribute__((ext_vector_type(2))) float v2f;
typedef __attribute__((ext_vector_type(8))) float v8f;

#define NC      100000
#define NU      100000
#define NEDGE   1600000
#define DC      256
#define HID     128
#define OUTD    64

// ---- WMMA f32 16x16x4 wrapper -------------------------------------------
__device__ __forceinline__ v8f wmma4(v2f a, v2f b, v8f c) {
  // 8 args: (neg_a, A, neg_b, B, c_mod, C, reuse_a, reuse_b)
  return __builtin_amdgcn_wmma_f32_16x16x4_f32(false, a, false, b, (short)0, c,
                                               false, false);
}

// A-fragment (16x4, f32) from a row-major [.., K] tile in LDS.
// ISA layout: VGPR0 = K = 2*half, VGPR1 = K = 2*half+1, rows on lanes 0..15.
__device__ __forceinline__ v2f afrag(const float* As, int K, int rowBase,
                                     int lane, int k0) {
  int r = lane & 15;
  int half = lane >> 4;
  const float* p = As + (rowBase + r) * K + k0 + 2 * half;
  return *(const v2f*)p;
}

// B-fragment (4x16, f32) from a row-major [K, N] weight matrix in global.
// Mirrors the A layout across the K dimension (compile-only; layout assumed).
__device__ __forceinline__ v2f bfrag(const float* W, int N, int k0, int n0,
                                     int lane) {
  int r = lane & 15;
  int half = lane >> 4;
  int k = k0 + 2 * half;
  v2f b;
  b.x = W[(size_t)k * N + n0 + r];
  b.y = W[(size_t)(k + 1) * N + n0 + r];
  return b;
}

// ---- K1/K2: per-node-type projection + relu ------------------------------
// Block = 256 threads = 8 waves. M-tile = 32 rows (2 row-tiles), N = 128
// (wave w owns column tile w). Grid = M/32 exactly (100000 % 32 == 0).
template <int K>
__global__ __launch_bounds__(256) void proj_relu_wmma(
    const float* __restrict__ X, const float* __restrict__ W,
    const float* __restrict__ bias, float* __restrict__ Hout) {
  __shared__ __align__(16) float As[32 * K];  // 16 KB (K=128) / 32 KB (K=256)
  const int tid = threadIdx.x;
  const int lane = tid & 31;
  const int wave = tid >> 5;
  const int m0 = blockIdx.x * 32;

  // Cooperative, coalesced stage of the 32xK input tile (float4 per thread).
  const int totalVecs = 32 * (K / 4);
  const float4* Xv = (const float4*)(X + (size_t)m0 * K);
  float4* Asv = (float4*)As;
  for (int i = tid; i < totalVecs; i += 256) Asv[i] = Xv[i];
  __syncthreads();

  const int n0 = wave * 16;
  v8f c0 = {};
  v8f c1 = {};
  for (int k0 = 0; k0 < K; k0 += 4) {
    v2f a0 = afrag(As, K, 0, lane, k0);
    v2f a1 = afrag(As, K, 16, lane, k0);
    v2f b = bfrag(W, HID, k0, n0, lane);  // shared by both row tiles
    c0 = wmma4(a0, b, c0);
    c1 = wmma4(a1, b, c1);
  }

  // C/D layout: VGPR v -> row (v + 8*half), col = lane&15 within the tile.
  const int r = lane & 15;
  const int half = lane >> 4;
  const float bv = bias[n0 + r];
  float* o = Hout + (size_t)m0 * HID;
#pragma unroll
  for (int v = 0; v < 8; ++v) {
    int row = v + 8 * half;
    o[(size_t)row * HID + n0 + r] = fmaxf(c0[v] + bv, 0.0f);
    o[(size_t)(16 + row) * HID + n0 + r] = fmaxf(c1[v] + bv, 0.0f);
  }
}

// ---- K3: zero the aggregation workspace ----------------------------------
__global__ void zero_f32(float* __restrict__ p, size_t n) {
  size_t i = (size_t)blockIdx.x * blockDim.x + threadIdx.x;
  if (i < n) p[i] = 0.0f;
}

// ---- K4: edge scatter (one wave32 per edge, 128 floats = 4 per lane) -----
__global__ __launch_bounds__(256) void edge_scatter(
    const float* __restrict__ h_u, const long long* __restrict__ ei,
    float* __restrict__ agg, float* __restrict__ cnt, int E) {
  int gtid = blockIdx.x * 256 + threadIdx.x;
  int edge = gtid >> 5;  // wave id == edge id
  int lane = threadIdx.x & 31;
  if (edge >= E) return;
  long long src = ei[edge];            // edge_index[0][e]
  long long dst = ei[(size_t)E + edge];  // edge_index[1][e]
  // h_u row is L2-resident (51 MB < 192 MB L2): gather + atomic-add to L2.
  const float4 v = ((const float4*)(h_u + (size_t)src * HID))[lane];
  float* a = agg + (size_t)dst * HID + lane * 4;
  atomicAdd(a + 0, v.x);
  atomicAdd(a + 1, v.y);
  atomicAdd(a + 2, v.z);
  atomicAdd(a + 3, v.w);
  if (lane == 0) atomicAdd(cnt + dst, 1.0f);
}

// ---- K5: fused mean-div + SAGE combine + output projection ---------------
__global__ __launch_bounds__(256) void sage_fused(
    const float* __restrict__ agg, const float* __restrict__ cnt,
    const float* __restrict__ h_c, const float* __restrict__ Wl,
    const float* __restrict__ bl, const float* __restrict__ Wr,
    const float* __restrict__ Wo, const float* __restrict__ bo,
    float* __restrict__ out) {
  __shared__ __align__(16) float Ms[32 * HID];  // mean tile
  __shared__ __align__(16) float Cs[32 * HID];  // h_c tile
  __shared__ __align__(16) float Hs[32 * HID];  // hidden result tile
  const int tid = threadIdx.x;
  const int lane = tid & 31;
  const int wave = tid >> 5;
  const int m0 = blockIdx.x * 32;

  // Stage mean (= agg / max(cnt,1)) and h_c tiles.
  const float4* av = (const float4*)(agg + (size_t)m0 * HID);
  const float4* cv = (const float4*)(h_c + (size_t)m0 * HID);
  float4* Msv = (float4*)Ms;
  float4* Csv = (float4*)Cs;
  for (int i = tid; i < 32 * (HID / 4); i += 256) {
    int row = i >> 5;  // HID/4 = 32 float4 per row
    float s = 1.0f / fmaxf(cnt[m0 + row], 1.0f);
    float4 a = av[i];
    a.x *= s; a.y *= s; a.z *= s; a.w *= s;
    Msv[i] = a;
    Csv[i] = cv[i];
  }
  __syncthreads();

  // Phase 1: h = mean @ Wl + h_c @ Wr + bl   (wave w -> column tile w)
  {
    const int n0 = wave * 16;
    v8f c0 = {};
    v8f c1 = {};
    for (int k0 = 0; k0 < HID; k0 += 4) {
      v2f am0 = afrag(Ms, HID, 0, lane, k0);
      v2f am1 = afrag(Ms, HID, 16, lane, k0);
      v2f bL = bfrag(Wl, HID, k0, n0, lane);
      c0 = wmma4(am0, bL, c0);
      c1 = wmma4(am1, bL, c1);
      v2f ac0 = afrag(Cs, HID, 0, lane, k0);
      v2f ac1 = afrag(Cs, HID, 16, lane, k0);
      v2f bR = bfrag(Wr, HID, k0, n0, lane);
      c0 = wmma4(ac0, bR, c0);
      c1 = wmma4(ac1, bR, c1);
    }
    const int r = lane & 15;
    const int half = lane >> 4;
    const float bv = bl[n0 + r];
#pragma unroll
    for (int v = 0; v < 8; ++v) {
      int row = v + 8 * half;
      Hs[row * HID + n0 + r] = c0[v] + bv;
      Hs[(16 + row) * HID + n0 + r] = c1[v] + bv;
    }
  }
  __syncthreads();

  // Phase 2: out = h @ Wo + bo.  8 tiles (2 row x 4 col) over 8 waves.
  {
    const int rt = wave >> 2;
    const int ct = wave & 3;
    const int n0 = ct * 16;
    v8f c = {};
    for (int k0 = 0; k0 < HID; k0 += 4) {
      v2f a = afrag(Hs, HID, rt * 16, lane, k0);
      v2f b = bfrag(Wo, OUTD, k0, n0, lane);
      c = wmma4(a, b, c);
    }
    const int r = lane & 15;
    const int half = lane >> 4;
    const float bv = bo[n0 + r];
    float* o = out + (size_t)(m0 + rt * 16) * OUTD;
#pragma unroll
    for (int v = 0; v < 8; ++v) {
      o[(size_t)(v + 8 * half) * OUTD + n0 + r] = c[v] + bv;
    }
  }
}

// ---------------------------------------------------------------------------
extern "C" void kernel_launch(void* const* d_in, const int* in_sizes, int n_in,
                              void* d_out, int out_size, void* d_ws,
                              size_t ws_size, hipStream_t stream) {
  const float* x_content = (const float*)d_in[0];
  const float* x_user = (const float*)d_in[1];
  const long long* edge_index = (const long long*)d_in[2];  // int64 [2, E]
  const float* Wc = (const float*)d_in[3];
  const float* bc = (const float*)d_in[4];
  const float* Wu = (const float*)d_in[5];
  const float* bu = (const float*)d_in[6];
  const float* Wl = (const float*)d_in[7];
  const float* bl = (const float*)d_in[8];
  const float* Wr = (const float*)d_in[9];
  const float* Wo = (const float*)d_in[10];
  const float* bo = (const float*)d_in[11];
  float* out = (float*)d_out;

  // Workspace layout (floats): h_c | h_u | agg | cnt  (~154 MB)
  float* h_c = (float*)d_ws;
  float* h_u = h_c + (size_t)NC * HID;
  float* agg = h_u + (size_t)NU * HID;
  float* cnt = agg + (size_t)NC * HID;

  // K3: zero agg + cnt (contiguous region).
  size_t nz = (size_t)NC * HID + NC;
  zero_f32<<<(unsigned)((nz + 255) / 256), 256, 0, stream>>>(agg, nz);

  // K1/K2: input projections (100000 % 32 == 0 -> no tail).
  proj_relu_wmma<DC><<<NC / 32, 256, 0, stream>>>(x_content, Wc, bc, h_c);
  proj_relu_wmma<HID><<<NU / 32, 256, 0, stream>>>(x_user, Wu, bu, h_u);

  // K4: scatter-sum over edges (one wave per edge).
  int scatterBlocks = (NEDGE * 32) / 256;  // 200000, exact
  edge_scatter<<<scatterBlocks, 256, 0, stream>>>(h_u, edge_index, agg, cnt,
                                                  NEDGE);

  // K5: fused mean + SAGE combine + output projection.
  sage_fused<<<NC / 32, 256, 0, stream>>>(agg, cnt, h_c, Wl, bl, Wr, Wo, bo,
                                          out);
}